// SpatialAdaptiveSynBatchNorm2d_part_65532611002801
// MI455X (gfx1250) — compile-verified
//
#include <hip/hip_runtime.h>
#include <hip/hip_bf16.h>

// ---------------------------------------------------------------------------
// SpatialAdaptiveSynBatchNorm2d fused implementation for MI455X (gfx1250).
//
// Shapes: b=8, o=16, c=128, h=w=128 (HW=16384), num_w=512.
// Bandwidth-bound overall (~150-220MB traffic, ~3.3 GFLOP) -> spatial part in
// fp32 VALU with coalesced loads; the dense parameter GEMM (128x384x512, fp32)
// uses V_WMMA_F32_16X16X4_F32 (CDNA5 pure-fp32 matrix op).
// ---------------------------------------------------------------------------

typedef __attribute__((ext_vector_type(2))) float v2f;
typedef __attribute__((ext_vector_type(8))) float v8f;

#define CC    128
#define HW    16384
#define BB    8
#define OO    16
#define NW    512
#define N3C   384            // 3*c
#define CHW   (CC * HW)      // per-batch x stride

// ---------------------------------------------------------------------------
// Kernel A: per-channel mean / rstd over (b, h, w).  One block per channel.
// ---------------------------------------------------------------------------
__global__ __launch_bounds__(256)
void bn_stats_kernel(const float* __restrict__ x,
                     float* __restrict__ meanv, float* __restrict__ rstdv)
{
    const int c = blockIdx.x;
    const int t = threadIdx.x;
    float s = 0.0f, s2 = 0.0f;
    // 8 batches x 16384 pixels per channel; float4 loads, lane-contiguous.
    for (int b = 0; b < BB; ++b) {
        const float4* p4 = (const float4*)(x + (size_t)b * CHW + (size_t)c * HW);
        for (int i = t; i < HW / 4; i += 256) {
            float4 v = p4[i];
            s  += v.x + v.y + v.z + v.w;
            s2 += v.x * v.x + v.y * v.y + v.z * v.z + v.w * v.w;
        }
    }
    __shared__ float rs[256], rs2[256];
    rs[t] = s; rs2[t] = s2;
    __syncthreads();
    for (int st = 128; st > 0; st >>= 1) {
        if (t < st) { rs[t] += rs[t + st]; rs2[t] += rs2[t + st]; }
        __syncthreads();
    }
    if (t == 0) {
        const float inv_n = 1.0f / (float)(BB * HW);
        float m = rs[0] * inv_n;
        float var = rs2[0] * inv_n - m * m;
        meanv[c] = m;
        rstdv[c] = rsqrtf(var + 1e-5f);
    }
}

// ---------------------------------------------------------------------------
// Kernel B: wa = vector @ wp_w^T + wp_b ; ba = vector @ bp_w^T + bp_b
// M=128, N=384, K=512, all fp32 -> V_WMMA_F32_16X16X4_F32.
// 384 wave-tiles total (8x24 tiles x 2 matrices); 48 blocks x 8 waves.
//
// ISA lane layouts (wave32):
//   A (16x4 f32):  lanes 0-15 hold row M=lane, VGPR0/1 = K0/K1;
//                  lanes 16-31 hold row M=lane-16, VGPR0/1 = K2/K3.
//   B (4x16 f32):  mirrored: lanes 0-15 col N=lane with K0/K1, lanes 16-31 K2/K3.
//   C/D (16x16):   VGPR i: lanes 0-15 -> M=i, lanes 16-31 -> M=i+8; N=lane%16.
// ---------------------------------------------------------------------------
__global__ __launch_bounds__(256)
void affine_gemm_kernel(const float* __restrict__ vec,
                        const float* __restrict__ wp_w, const float* __restrict__ wp_b,
                        const float* __restrict__ bp_w, const float* __restrict__ bp_b,
                        float* __restrict__ wa, float* __restrict__ ba)
{
    const int wave = (blockIdx.x * blockDim.x + threadIdx.x) >> 5; // 0..383
    const int lane = threadIdx.x & 31;
    const int sel  = wave / 192;          // 0 -> wa, 1 -> ba
    const int tile = wave % 192;
    const int mt   = tile & 7;            // 8 M-tiles
    const int nt   = tile >> 3;           // 24 N-tiles

    const float* Wm   = sel ? bp_w : wp_w;
    const float* bias = sel ? bp_b : wp_b;
    float*       dst  = sel ? ba   : wa;

    const int half = lane >> 4;           // 0: K0/K1, 1: K2/K3
    const int l16  = lane & 15;
    const float* aptr = vec + (size_t)(mt * 16 + l16) * NW + 2 * half;
    const float* bptr = Wm  + (size_t)(nt * 16 + l16) * NW + 2 * half;

    v8f acc = {};
    for (int kk = 0; kk < NW; kk += 4) {
        v2f a, bv;
        a.x  = aptr[kk]; a.y  = aptr[kk + 1];
        bv.x = bptr[kk]; bv.y = bptr[kk + 1];
        // D = A x B + C  (16x16x4, pure fp32)
        acc = __builtin_amdgcn_wmma_f32_16x16x4_f32(
            false, a, false, bv, (short)0, acc, false, false);
    }

    const int cidx = nt * 16 + l16;
    const float bia = bias[cidx];
#pragma unroll
    for (int i = 0; i < 8; ++i) {
        const int r = mt * 16 + half * 8 + i;
        dst[(size_t)r * N3C + cidx] = acc[i] + bia;
    }
}

// ---------------------------------------------------------------------------
// Kernel C: fused spatial einsum + BN apply.
// Block = 256 threads = 8 waves. Tile = 32 pixels, all 128 channels, 1 batch.
//   lane  = pixel within tile (coalesced x/bbox/out accesses)
//   wave  = channel group of 16
// LDS: per-batch affine params (2 x 24KB), normalized bboxes (6KB), inv-den.
// Algebra: W0+W1+W2 = sum_{k,o} (bb_k/den_k) * w_k  + 3   (same for B, +0)
// ---------------------------------------------------------------------------
__global__ __launch_bounds__(256)
void fused_spatial_kernel(const float* __restrict__ x,
                          const float* __restrict__ bb0,
                          const float* __restrict__ bb1,
                          const float* __restrict__ bb2,
                          const float* __restrict__ wa,
                          const float* __restrict__ ba,
                          const float* __restrict__ meanv,
                          const float* __restrict__ rstdv,
                          float* __restrict__ out)
{
    __shared__ float sW[OO * N3C];     // [o][k][c] = 6144 floats (24KB)
    __shared__ float sB[OO * N3C];     // 6144 floats (24KB)
    __shared__ float sBB[3 * OO * 32]; // [k][o][p] = 1536 floats (6KB)
    __shared__ float sInv[3 * 32];     // [k][p]

    const int t       = threadIdx.x;
    const int b       = blockIdx.x >> 9;          // 512 pixel-tiles per batch
    const int pixbase = (blockIdx.x & 511) * 32;

    // Stage per-batch affine params: wa rows b*16..b*16+15 are contiguous,
    // and [row][k*128+c] == [o][k][c], so this is a flat 6144-float copy.
    const float* waB = wa + (size_t)b * OO * N3C;
    const float* baB = ba + (size_t)b * OO * N3C;
    for (int i = t; i < OO * N3C; i += 256) { sW[i] = waB[i]; sB[i] = baB[i]; }

    // Stage bbox tiles: sBB[k*512 + o*32 + p], lane-contiguous in p.
    for (int i = t; i < OO * 32; i += 256) {
        const int o = i >> 5, p = i & 31;
        const size_t g = (size_t)(b * OO + o) * HW + pixbase + p;
        sBB[i]        = bb0[g];
        sBB[512 + i]  = bb1[g];
        sBB[1024 + i] = bb2[g];
    }
    __syncthreads();

    // Per-pixel denominators (one per bbox variant).
    if (t < 96) {
        const int k = t >> 5, p = t & 31;
        float s = 0.0f;
#pragma unroll
        for (int o = 0; o < OO; ++o) s += sBB[k * 512 + o * 32 + p];
        sInv[t] = 1.0f / (s + 1e-6f);
    }
    __syncthreads();

    // Normalize bboxes in place -> the 3 variants merge into one reduction.
    for (int i = t; i < 3 * OO * 32; i += 256) {
        const int k = i >> 9, p = i & 31;
        sBB[i] *= sInv[k * 32 + p];
    }
    __syncthreads();

    const int lane  = t & 31;      // pixel
    const int cbase = (t >> 5) * 16;

    float accW[16], accB[16];
#pragma unroll
    for (int i = 0; i < 16; ++i) { accW[i] = 0.0f; accB[i] = 0.0f; }

    for (int o = 0; o < OO; ++o) {
        const float a0 = sBB[o * 32 + lane];
        const float a1 = sBB[512 + o * 32 + lane];
        const float a2 = sBB[1024 + o * 32 + lane];
        const float* wro = sW + o * N3C + cbase;   // [k*128 + c] broadcasts
        const float* bro = sB + o * N3C + cbase;
#pragma unroll
        for (int i = 0; i < 16; ++i) {
            accW[i] += a0 * wro[i] + a1 * wro[128 + i] + a2 * wro[256 + i];
            accB[i] += a0 * bro[i] + a1 * bro[128 + i] + a2 * bro[256 + i];
        }
    }

    // BN apply + affine; x/out accesses coalesced along pixels.
    const int pix = pixbase + lane;
#pragma unroll
    for (int i = 0; i < 16; ++i) {
        const int c = cbase + i;
        const size_t idx = (size_t)(b * CC + c) * HW + pix;
        const float xn = (x[idx] - meanv[c]) * rstdv[c];
        out[idx] = (accW[i] + 3.0f) * xn + accB[i];
    }
}

// ---------------------------------------------------------------------------
extern "C" void kernel_launch(void* const* d_in, const int* in_sizes, int n_in,
                              void* d_out, int out_size, void* d_ws, size_t ws_size,
                              hipStream_t stream) {
    const float* x      = (const float*)d_in[0];
    const float* vec    = (const float*)d_in[1];
    const float* bbox0  = (const float*)d_in[2];
    const float* bbox1  = (const float*)d_in[3];
    const float* bbox2  = (const float*)d_in[4];
    const float* wp_w   = (const float*)d_in[5];
    const float* wp_b   = (const float*)d_in[6];
    const float* bp_w   = (const float*)d_in[7];
    const float* bp_b   = (const float*)d_in[8];
    float* out = (float*)d_out;

    // Workspace layout (floats): mean[128] | rstd[128] | wa[128*384] | ba[128*384]
    float* ws    = (float*)d_ws;
    float* meanv = ws;
    float* rstdv = ws + 128;
    float* wa    = ws + 256;
    float* ba    = ws + 256 + 128 * N3C;

    bn_stats_kernel<<<CC, 256, 0, stream>>>(x, meanv, rstdv);
    affine_gemm_kernel<<<48, 256, 0, stream>>>(vec, wp_w, wp_b, bp_w, bp_b, wa, ba);
    fused_spatial_kernel<<<BB * (HW / 32), 256, 0, stream>>>(
        x, bbox0, bbox1, bbox2, wa, ba, meanv, rstdv, out);
}